// DecoderLayer_84164179133306
// MI455X (gfx1250) — compile-verified
//
#include <hip/hip_runtime.h>
#include <hip/hip_bf16.h>

// ---------------- problem constants ----------------
constexpr int Bb  = 16;
constexpr int Ls  = 512;
constexpr int Ss  = 1024;
constexpr int Dd  = 512;
constexpr int Hh  = 8;
constexpr int Mm  = 64;     // fourier modes kept
constexpr int DFFf= 2048;
constexpr int COo = 512;
constexpr int DK  = 64;     // D / H
constexpr int DH  = 256;    // D / 2 (gating hidden)

typedef __attribute__((ext_vector_type(2))) float v2f;
typedef __attribute__((ext_vector_type(8))) float v8f;

static __device__ __forceinline__ v8f wmma4(v2f a, v2f b, v8f c) {
  // V_WMMA_F32_16X16X4_F32 : D = A(16x4,f32) * B(4x16,f32) + C(16x16,f32)
  return __builtin_amdgcn_wmma_f32_16x16x4_f32(false, a, false, b, (short)0, c,
                                               false, false);
}

static __device__ __forceinline__ v8f v8zero() {
  v8f z = {0.f,0.f,0.f,0.f,0.f,0.f,0.f,0.f};
  return z;
}

// ---------------- DFT / iDFT tables ----------------
// qft[m] = sum_l x[l] * (cos(2pi m l / L) - i sin(2pi m l / L))
// irfft(64-mode spectrum)[l] = sum_m w_m/L * (Xr cos(th) - Xi sin(th)),
//   w_0 = 1, w_m = 2 (M < L/2 so no Nyquist term)
__global__ void init_tables_kernel(float* FcL, float* FsL, float* FcS,
                                   float* FsS, float* ILc, float* ILs) {
  const int i = blockIdx.x * blockDim.x + threadIdx.x;
  const double TWO_PI = 6.283185307179586476925286766559;
  if (i < Mm * Ls) {
    int m = i / Ls, l = i % Ls;
    double ang = -TWO_PI * (double)(((long)m * l) % Ls) / (double)Ls;
    FcL[i] = (float)cos(ang);
    FsL[i] = (float)sin(ang);
  }
  if (i < Mm * Ss) {
    int m = i / Ss, l = i % Ss;
    double ang = -TWO_PI * (double)(((long)m * l) % Ss) / (double)Ss;
    FcS[i] = (float)cos(ang);
    FsS[i] = (float)sin(ang);
  }
  if (i < Ls * Mm) {
    int l = i / Mm, m = i % Mm;
    double w = (m == 0) ? 1.0 : 2.0;
    double ang = TWO_PI * (double)(((long)m * l) % Ls) / (double)Ls;
    ILc[i] = (float)( w * cos(ang) / (double)Ls);
    ILs[i] = (float)(-w * sin(ang) / (double)Ls);
  }
}

// ---------------- generic GEMM: C = A[M,K] @ W[N,K]^T (+bias)(+relu)(+res) --
// Block = 8 waves: one shared 16-row A panel, 8 distinct 16-col B tiles
// (128 cols). Both panels staged in LDS (row stride padded 64->68 floats so
// the 16 fragment rows land in distinct bank groups).
template <int RELU, int HASRES>
__global__ __launch_bounds__(256) void gemm_wt_kernel(
    const float* __restrict__ A, const float* __restrict__ W,
    const float* __restrict__ bias, const float* __restrict__ Res,
    float* __restrict__ C, int Mr, int N, int K) {
  (void)Mr;
  constexpr int KT  = 64;
  constexpr int LDP = KT + 4;           // bank-conflict padding
  __shared__ float As[16 * LDP];        //  ~4.3 KB
  __shared__ float Bs[128 * LDP];       // ~34.8 KB
  const int tid  = threadIdx.x;
  const int lane = tid & 31;
  const int wave = tid >> 5;
  const int half = lane >> 4;           // K-pair select within the 4-wide step
  const int lr   = lane & 15;           // A row / B col / C col
  const int nB = blockIdx.x * 128;      // block column base
  const int n0 = nB + wave * 16;
  const int m0 = blockIdx.y * 16;
  v8f acc = v8zero();
  for (int kb = 0; kb < K; kb += KT) {
    if (kb + KT < K) {                  // keep next chunk warm in GL2/L0
      __builtin_prefetch(&A[(size_t)(m0 + (tid & 15)) * K + kb + KT], 0, 1);
      __builtin_prefetch(&W[(size_t)(nB + (tid >> 1)) * K + kb + KT], 0, 1);
    }
    // A panel 16 x KT, coalesced: consecutive tid -> consecutive k
    for (int t = tid; t < 16 * KT; t += 256) {
      const int row = t >> 6, kk = t & (KT - 1);
      As[row * LDP + kk] = A[(size_t)(m0 + row) * K + kb + kk];
    }
    // B panel 128 x KT from W[N,K]
    for (int t = tid; t < 128 * KT; t += 256) {
      const int row = t >> 6, kk = t & (KT - 1);
      Bs[row * LDP + kk] = W[(size_t)(nB + row) * K + kb + kk];
    }
    __syncthreads();
    const float* Aw = &As[lr * LDP];
    const float* Bw = &Bs[(wave * 16 + lr) * LDP];
    for (int k0 = 0; k0 < KT; k0 += 4) {
      const int kk = k0 + half * 2;
      v2f a; a.x = Aw[kk]; a.y = Aw[kk + 1];
      v2f b; b.x = Bw[kk]; b.y = Bw[kk + 1];
      acc = wmma4(a, b, acc);
    }
    __syncthreads();
  }
  const int r0 = half * 8;
  const float bv = bias ? bias[n0 + lr] : 0.f;
  for (int r = 0; r < 8; ++r) {
    const size_t idx = (size_t)(m0 + r0 + r) * N + n0 + lr;
    float v = acc[r] + bv;
    if (RELU) v = fmaxf(v, 0.f);
    if (HASRES) v += Res[idx];
    C[idx] = v;
  }
}

// ---------------- forward DFT: dst[z,m,n] = sum_l T[m,l]*src(z)[l,n] --------
// z = b*Hn + h ; src element (l,n) at src + b*batchStride + h*headStride
//                                     + l*rowStride + n
__global__ __launch_bounds__(128) void dft_fwd_kernel(
    const float* __restrict__ Tc, const float* __restrict__ Ts,
    const float* __restrict__ src, int Hn, int headStride, long long batchStride,
    int rowStride, int K, int N, float* __restrict__ dstR,
    float* __restrict__ dstI) {
  const int lane = threadIdx.x & 31;
  const int wave = threadIdx.x >> 5;
  const int half = lane >> 4;
  const int lr   = lane & 15;
  const int n0 = (blockIdx.x * 4 + wave) * 16;
  const int m0 = blockIdx.y * 16;
  const int z  = blockIdx.z;
  const int b = z / Hn, h = z % Hn;
  const float* S   = src + (size_t)b * batchStride + (size_t)h * headStride;
  const float* Tcr = Tc + (size_t)(m0 + lr) * K;
  const float* Tsr = Ts + (size_t)(m0 + lr) * K;
  v8f ar = v8zero(), ai = v8zero();
  for (int k0 = 0; k0 < K; k0 += 4) {
    const int kk = k0 + half * 2;
    v2f bf; bf.x = S[(size_t)kk * rowStride + n0 + lr];
            bf.y = S[(size_t)(kk + 1) * rowStride + n0 + lr];
    v2f ac; ac.x = Tcr[kk]; ac.y = Tcr[kk + 1];
    v2f as; as.x = Tsr[kk]; as.y = Tsr[kk + 1];
    ar = wmma4(ac, bf, ar);
    ai = wmma4(as, bf, ai);
  }
  const int r0 = half * 8;
  for (int r = 0; r < 8; ++r) {
    const size_t idx = (size_t)z * Mm * N + (size_t)(m0 + r0 + r) * N + n0 + lr;
    dstR[idx] = ar[r];
    dstI[idx] = ai[r];
  }
}

// ---------------- FEB per-mode complex mix: om[b,:,e] = q[b,:,.] @ W[.,e,m] --
// W is read with stride M (layout [D,D,M]); the full 134 MB tensor fits the
// 192 MB global L2 so every mode after the first hits cache.
__global__ __launch_bounds__(128) void feb_mode_kernel(
    const float* __restrict__ qr, const float* __restrict__ qi,
    const float* __restrict__ wr, const float* __restrict__ wi,
    float* __restrict__ omr, float* __restrict__ omi) {
  const int lane = threadIdx.x & 31;
  const int wave = threadIdx.x >> 5;
  const int half = lane >> 4;
  const int lr   = lane & 15;
  const int m  = blockIdx.y;                    // mode
  const int n0 = (blockIdx.x * 4 + wave) * 16;  // e tile
  const float* Aq = qr + (size_t)lr * Mm * Dd + (size_t)m * Dd;  // row = batch
  const float* Ai = qi + (size_t)lr * Mm * Dd + (size_t)m * Dd;
  const size_t wb = (size_t)(n0 + lr) * Mm + m;  // W[d,e,m]: + d*Dd*Mm
  v8f aRR = v8zero(), aII = v8zero(), aRI = v8zero(), aIR = v8zero();
  for (int k0 = 0; k0 < Dd; k0 += 4) {
    const int kk = k0 + half * 2;
    v2f axr; axr.x = Aq[kk]; axr.y = Aq[kk + 1];
    v2f axi; axi.x = Ai[kk]; axi.y = Ai[kk + 1];
    v2f bwr; bwr.x = wr[(size_t)kk * Dd * Mm + wb];
             bwr.y = wr[(size_t)(kk + 1) * Dd * Mm + wb];
    v2f bwi; bwi.x = wi[(size_t)kk * Dd * Mm + wb];
             bwi.y = wi[(size_t)(kk + 1) * Dd * Mm + wb];
    aRR = wmma4(axr, bwr, aRR);
    aII = wmma4(axi, bwi, aII);
    aRI = wmma4(axr, bwi, aRI);
    aIR = wmma4(axi, bwr, aIR);
  }
  const int r0 = half * 8;
  for (int r = 0; r < 8; ++r) {
    const size_t idx =
        (size_t)(r0 + r) * Mm * Dd + (size_t)m * Dd + n0 + lr;  // [B,M,D]
    omr[idx] = aRR[r] - aII[r];
    omi[idx] = aRI[r] + aIR[r];
  }
}

// ---------------- inverse DFT: dst(z)[l, n] = sum_m ILc[l,m]Xr + ILs[l,m]Xi -
// z = b*Hn + h ; dst element at dst + b*Ls*dstStride + l*dstStride + h*N + n
__global__ __launch_bounds__(128) void irfft_kernel(
    const float* __restrict__ ILc, const float* __restrict__ ILs,
    const float* __restrict__ srcR, const float* __restrict__ srcI, int N,
    int Hn, float* __restrict__ dst, int dstStride,
    const float* __restrict__ res) {
  const int lane = threadIdx.x & 31;
  const int wave = threadIdx.x >> 5;
  const int half = lane >> 4;
  const int lr   = lane & 15;
  const int n0 = (blockIdx.x * 4 + wave) * 16;
  const int l0 = blockIdx.y * 16;
  const int z  = blockIdx.z;
  const int b = z / Hn, h = z % Hn;
  const float* Ac = ILc + (size_t)(l0 + lr) * Mm;
  const float* As = ILs + (size_t)(l0 + lr) * Mm;
  const float* SR = srcR + (size_t)z * Mm * N;
  const float* SI = srcI + (size_t)z * Mm * N;
  v8f acc = v8zero();
  for (int k0 = 0; k0 < Mm; k0 += 4) {
    const int kk = k0 + half * 2;
    v2f ac; ac.x = Ac[kk]; ac.y = Ac[kk + 1];
    v2f as; as.x = As[kk]; as.y = As[kk + 1];
    v2f br; br.x = SR[(size_t)kk * N + n0 + lr];
            br.y = SR[(size_t)(kk + 1) * N + n0 + lr];
    v2f bi; bi.x = SI[(size_t)kk * N + n0 + lr];
            bi.y = SI[(size_t)(kk + 1) * N + n0 + lr];
    acc = wmma4(ac, br, acc);
    acc = wmma4(as, bi, acc);
  }
  const int r0 = half * 8;
  for (int r = 0; r < 8; ++r) {
    const size_t idx = (size_t)b * Ls * dstStride +
                       (size_t)(l0 + r0 + r) * dstStride + (size_t)h * N + n0 + lr;
    float v = acc[r];
    if (res) v += res[idx];
    dst[idx] = v;
  }
}

// ---------------- frequency-domain attention ----------------
__global__ __launch_bounds__(128) void attn_logits_kernel(
    const float* __restrict__ qfr, const float* __restrict__ qfi,
    const float* __restrict__ kfr, const float* __restrict__ kfi,
    float* __restrict__ attn) {
  const int lane = threadIdx.x & 31;
  const int wave = threadIdx.x >> 5;
  const int half = lane >> 4;
  const int lr   = lane & 15;
  const int z  = blockIdx.z;            // b*H + h
  const int l0 = blockIdx.y * 16;
  const int n0 = wave * 16;             // s tile (4 waves cover 64)
  const size_t base = (size_t)z * 64 * 64;
  const float* Qr = qfr + base + (size_t)(l0 + lr) * 64;
  const float* Qi = qfi + base + (size_t)(l0 + lr) * 64;
  const float* Kr = kfr + base + (size_t)(n0 + lr) * 64;
  const float* Ki = kfi + base + (size_t)(n0 + lr) * 64;
  v8f acc = v8zero();
  for (int k0 = 0; k0 < 64; k0 += 4) {
    const int kk = k0 + half * 2;
    v2f aqr; aqr.x = Qr[kk]; aqr.y = Qr[kk + 1];
    v2f aqi; aqi.x = Qi[kk]; aqi.y = Qi[kk + 1];
    v2f bkr; bkr.x = Kr[kk]; bkr.y = Kr[kk + 1];
    v2f bki; bki.x = Ki[kk]; bki.y = Ki[kk + 1];
    acc = wmma4(aqr, bkr, acc);
    acc = wmma4(aqi, bki, acc);
  }
  const int r0 = half * 8;
  for (int r = 0; r < 8; ++r)
    attn[base + (size_t)(l0 + r0 + r) * 64 + n0 + lr] = acc[r];
}

__global__ __launch_bounds__(256) void softmax64_kernel(float* __restrict__ attn,
                                                        int rows) {
  const int row = blockIdx.x * 8 + (threadIdx.x >> 5);
  const int lane = threadIdx.x & 31;
  if (row >= rows) return;
  float* p = attn + (size_t)row * 64;
  float a = p[lane], b = p[lane + 32];
  float mx = fmaxf(a, b);
  for (int o = 16; o > 0; o >>= 1) mx = fmaxf(mx, __shfl_xor(mx, o, 32));
  float ea = __expf(a - mx), eb = __expf(b - mx);
  float sm = ea + eb;
  for (int o = 16; o > 0; o >>= 1) sm += __shfl_xor(sm, o, 32);
  p[lane]      = ea / sm;
  p[lane + 32] = eb / sm;
}

__global__ __launch_bounds__(128) void attn_apply_kernel(
    const float* __restrict__ attn, const float* __restrict__ vfr,
    const float* __restrict__ vfi, float* __restrict__ ofr,
    float* __restrict__ ofi) {
  const int lane = threadIdx.x & 31;
  const int wave = threadIdx.x >> 5;
  const int half = lane >> 4;
  const int lr   = lane & 15;
  const int z  = blockIdx.z;
  const int l0 = blockIdx.y * 16;
  const int n0 = wave * 16;             // dk tile
  const size_t base = (size_t)z * 64 * 64;
  const float* Ar = attn + base + (size_t)(l0 + lr) * 64;
  v8f accR = v8zero(), accI = v8zero();
  for (int k0 = 0; k0 < 64; k0 += 4) {
    const int kk = k0 + half * 2;
    v2f a; a.x = Ar[kk]; a.y = Ar[kk + 1];
    v2f br; br.x = vfr[base + (size_t)kk * 64 + n0 + lr];
            br.y = vfr[base + (size_t)(kk + 1) * 64 + n0 + lr];
    v2f bi; bi.x = vfi[base + (size_t)kk * 64 + n0 + lr];
            bi.y = vfi[base + (size_t)(kk + 1) * 64 + n0 + lr];
    accR = wmma4(a, br, accR);
    accI = wmma4(a, bi, accI);
  }
  const int r0 = half * 8;
  for (int r = 0; r < 8; ++r) {
    const size_t idx = base + (size_t)(l0 + r0 + r) * 64 + n0 + lr;
    ofr[idx] = accR[r];
    ofi[idx] = accI[r];
  }
}

// ---------------- MoE decomposition pieces ----------------
__global__ void pool_kernel(const float* __restrict__ x, float* __restrict__ p3,
                            float* __restrict__ p5, float* __restrict__ p7) {
  const size_t i = (size_t)blockIdx.x * blockDim.x + threadIdx.x;
  const size_t total = (size_t)Bb * Ls * Dd;
  if (i >= total) return;
  const int d = (int)(i % Dd);
  const size_t bl = i / Dd;
  const int l = (int)(bl % Ls);
  const size_t b = bl / Ls;
  const float* base = x + b * (size_t)Ls * Dd + d;
  float s3 = 0.f, s5 = 0.f, s7 = 0.f;
  for (int j = -3; j <= 3; ++j) {
    const int lj = l + j;
    if (lj < 0 || lj >= Ls) continue;       // zero "same" padding, fixed 1/k
    const float v = base[(size_t)lj * Dd];
    if (j >= -1 && j <= 1) s3 += v;
    if (j >= -2 && j <= 2) s5 += v;
    s7 += v;
  }
  p3[i] = s3 * (1.f / 3.f);
  p5[i] = s5 * (1.f / 5.f);
  p7[i] = s7 * (1.f / 7.f);
}

__global__ void gate_kernel(const float* __restrict__ gh,
                            const float* __restrict__ w2,
                            const float* __restrict__ b2, float* __restrict__ g,
                            int rows, int Kh) {
  const int r = blockIdx.x * blockDim.x + threadIdx.x;
  if (r >= rows) return;
  const float* hrow = gh + (size_t)r * Kh;
  float l0 = b2[0], l1 = b2[1], l2 = b2[2];
  for (int k = 0; k < Kh; ++k) {
    const float h = hrow[k];
    l0 += h * w2[k];
    l1 += h * w2[Kh + k];
    l2 += h * w2[2 * Kh + k];
  }
  const float mx = fmaxf(l0, fmaxf(l1, l2));
  const float e0 = __expf(l0 - mx), e1 = __expf(l1 - mx), e2 = __expf(l2 - mx);
  const float inv = 1.f / (e0 + e1 + e2);
  g[(size_t)r * 3 + 0] = e0 * inv;
  g[(size_t)r * 3 + 1] = e1 * inv;
  g[(size_t)r * 3 + 2] = e2 * inv;
}

__global__ void combine_kernel(const float* __restrict__ xin,
                               const float* __restrict__ p3,
                               const float* __restrict__ p5,
                               const float* __restrict__ p7,
                               const float* __restrict__ g,
                               float* __restrict__ xout,
                               float* __restrict__ trend) {
  const size_t i = (size_t)blockIdx.x * blockDim.x + threadIdx.x;
  const size_t total = (size_t)Bb * Ls * Dd;
  if (i >= total) return;
  const size_t row = i / Dd;
  const float t = g[row * 3 + 0] * p3[i] + g[row * 3 + 1] * p5[i] +
                  g[row * 3 + 2] * p7[i];
  trend[i] = t;
  xout[i]  = xin[i] - t;
}

// ---------------- circular conv of the 3 trends (3 taps = shifted GEMMs) ----
__global__ __launch_bounds__(256) void circconv_kernel(
    const float* __restrict__ t1, const float* __restrict__ t2,
    const float* __restrict__ t3, const float* __restrict__ w1,
    const float* __restrict__ w2, const float* __restrict__ w3,
    float* __restrict__ out) {
  const int lane = threadIdx.x & 31;
  const int wave = threadIdx.x >> 5;
  const int half = lane >> 4;
  const int lr   = lane & 15;
  const int n0 = (blockIdx.x * 8 + wave) * 16;  // CO tile
  const int l0 = blockIdx.y * 16;
  const int b  = blockIdx.z;
  const float* T[3] = {t1, t2, t3};
  const float* P[3] = {w1, w2, w3};
  v8f acc = v8zero();
  for (int j = 0; j < 3; ++j) {
    const float* Tb = T[j] + (size_t)b * Ls * Dd;
    const float* Pb = P[j];
    for (int t = 0; t < 3; ++t) {
      int lrow = l0 + lr + t - 1;
      lrow = (lrow + Ls) % Ls;                        // circular padding
      const float* Ar = Tb + (size_t)lrow * Dd;
      const float* Wc = Pb + (size_t)(n0 + lr) * Dd * 3 + t;  // p[o,d,t]
      for (int k0 = 0; k0 < Dd; k0 += 4) {
        const int kk = k0 + half * 2;
        v2f a; a.x = Ar[kk]; a.y = Ar[kk + 1];
        v2f bb; bb.x = Wc[(size_t)kk * 3]; bb.y = Wc[(size_t)(kk + 1) * 3];
        acc = wmma4(a, bb, acc);
      }
    }
  }
  const int r0 = half * 8;
  for (int r = 0; r < 8; ++r)
    out[(size_t)b * Ls * COo + (size_t)(l0 + r0 + r) * COo + n0 + lr] = acc[r];
}

// ---------------- workspace layout (floats) ----------------
constexpr long long F_FcL = 0;
constexpr long long F_FsL = F_FcL + (long long)Mm * Ls;
constexpr long long F_FcS = F_FsL + (long long)Mm * Ls;
constexpr long long F_FsS = F_FcS + (long long)Mm * Ss;
constexpr long long F_ILc = F_FsS + (long long)Mm * Ss;
constexpr long long F_ILs = F_ILc + (long long)Ls * Mm;
constexpr long long F_qr  = F_ILs + (long long)Ls * Mm;
constexpr long long F_qi  = F_qr  + (long long)Bb * Mm * Dd;
constexpr long long F_omr = F_qi  + (long long)Bb * Mm * Dd;
constexpr long long F_omi = F_omr + (long long)Bb * Mm * Dd;
constexpr long long F_qfr = F_omi + (long long)Bb * Mm * Dd;
constexpr long long F_qfi = F_qfr + (long long)Bb * Hh * Mm * DK;
constexpr long long F_kfr = F_qfi + (long long)Bb * Hh * Mm * DK;
constexpr long long F_kfi = F_kfr + (long long)Bb * Hh * Mm * DK;
constexpr long long F_vfr = F_kfi + (long long)Bb * Hh * Mm * DK;
constexpr long long F_vfi = F_vfr + (long long)Bb * Hh * Mm * DK;
constexpr long long F_att = F_vfi + (long long)Bb * Hh * Mm * DK;
constexpr long long F_ofr = F_att + (long long)Bb * Hh * Mm * Mm;
constexpr long long F_ofi = F_ofr + (long long)Bb * Hh * Mm * DK;
constexpr long long F_gh  = F_ofi + (long long)Bb * Hh * Mm * DK;
constexpr long long F_gts = F_gh  + (long long)Bb * Ls * DH;
constexpr long long F_xc  = F_gts + 32768;                    // gates (padded)
constexpr long long F_p3  = F_xc  + (long long)Bb * Ls * Dd;
constexpr long long F_p5  = F_p3  + (long long)Bb * Ls * Dd;
constexpr long long F_p7  = F_p5  + (long long)Bb * Ls * Dd;
constexpr long long F_t1  = F_p7  + (long long)Bb * Ls * Dd;
constexpr long long F_t2  = F_t1  + (long long)Bb * Ls * Dd;
constexpr long long F_t3  = F_t2  + (long long)Bb * Ls * Dd;
constexpr long long F_qb  = F_t3  + (long long)Bb * Ls * Dd;  // q proj, later octx
constexpr long long F_kv  = F_qb  + (long long)Bb * Ls * Dd;  // kp|vp, later ffn hidden

extern "C" void kernel_launch(void* const* d_in, const int* in_sizes, int n_in,
                              void* d_out, int out_size, void* d_ws,
                              size_t ws_size, hipStream_t stream) {
  (void)in_sizes; (void)n_in; (void)out_size; (void)ws_size;
  const float* x      = (const float*)d_in[0];
  const float* cross  = (const float*)d_in[1];
  const float* feb_wr = (const float*)d_in[2];
  const float* feb_wi = (const float*)d_in[3];
  const float* wq = (const float*)d_in[4];  const float* bq = (const float*)d_in[5];
  const float* wk = (const float*)d_in[6];  const float* bk = (const float*)d_in[7];
  const float* wv = (const float*)d_in[8];  const float* bv = (const float*)d_in[9];
  const float* wo = (const float*)d_in[10]; const float* bo = (const float*)d_in[11];
  const float* w_ff1 = (const float*)d_in[12];
  const float* w_ff2 = (const float*)d_in[13];
  const float* dw1[3] = {(const float*)d_in[14], (const float*)d_in[18], (const float*)d_in[22]};
  const float* db1[3] = {(const float*)d_in[15], (const float*)d_in[19], (const float*)d_in[23]};
  const float* dw2[3] = {(const float*)d_in[16], (const float*)d_in[20], (const float*)d_in[24]};
  const float* db2[3] = {(const float*)d_in[17], (const float*)d_in[21], (const float*)d_in[25]};
  const float* pc1 = (const float*)d_in[26];
  const float* pc2 = (const float*)d_in[27];
  const float* pc3 = (const float*)d_in[28];

  float* wsf = (float*)d_ws;
  float* FcL = wsf + F_FcL; float* FsL = wsf + F_FsL;
  float* FcS = wsf + F_FcS; float* FsS = wsf + F_FsS;
  float* ILc = wsf + F_ILc; float* ILs = wsf + F_ILs;
  float* qr  = wsf + F_qr;  float* qi  = wsf + F_qi;
  float* omr = wsf + F_omr; float* omi = wsf + F_omi;
  float* qfr = wsf + F_qfr; float* qfi = wsf + F_qfi;
  float* kfr = wsf + F_kfr; float* kfi = wsf + F_kfi;
  float* vfr = wsf + F_vfr; float* vfi = wsf + F_vfi;
  float* att = wsf + F_att; float* ofr = wsf + F_ofr; float* ofi = wsf + F_ofi;
  float* gh  = wsf + F_gh;  float* gts = wsf + F_gts;
  float* xc  = wsf + F_xc;
  float* p3b = wsf + F_p3;  float* p5b = wsf + F_p5;  float* p7b = wsf + F_p7;
  float* t1  = wsf + F_t1;  float* t2  = wsf + F_t2;  float* t3  = wsf + F_t3;
  float* qb  = wsf + F_qb;                      // q projection / attn context
  float* kp  = wsf + F_kv;                      // k projection
  float* vp  = wsf + F_kv + (long long)Bb * Ss * Dd;  // v projection
  float* ffh = wsf + F_kv;                      // FFN hidden (kp/vp dead)

  float* out_x  = (float*)d_out;
  float* out_tr = (float*)d_out + (size_t)Bb * Ls * Dd;

  const int BL  = Bb * Ls;                      // 8192
  const int BS  = Bb * Ss;                      // 16384
  const int ELT = Bb * Ls * Dd;                 // 4194304

  // --- DFT / iDFT coefficient tables (recomputed each call; deterministic) --
  init_tables_kernel<<<256, 256, 0, stream>>>(FcL, FsL, FcS, FsS, ILc, ILs);

  // --- Fourier enhanced block: x = x + irfft(einsum(rfft(x)[:M], W)) --------
  dft_fwd_kernel<<<dim3(Dd / 64, Mm / 16, Bb), 128, 0, stream>>>(
      FcL, FsL, x, 1, 0, (long long)Ls * Dd, Dd, Ls, Dd, qr, qi);
  feb_mode_kernel<<<dim3(Dd / 64, Mm), 128, 0, stream>>>(qr, qi, feb_wr, feb_wi,
                                                         omr, omi);
  irfft_kernel<<<dim3(Dd / 64, Ls / 16, Bb), 128, 0, stream>>>(
      ILc, ILs, omr, omi, Dd, 1, xc, Dd, x);

  // --- MoE decomposition (shared for the 3 call sites) ----------------------
  auto decomp = [&](int j, const float* xin, float* xout, float* trend) {
    pool_kernel<<<ELT / 256, 256, 0, stream>>>(xin, p3b, p5b, p7b);
    gemm_wt_kernel<1, 0><<<dim3(DH / 128, BL / 16), 256, 0, stream>>>(
        xin, dw1[j], db1[j], nullptr, gh, BL, DH, Dd);
    gate_kernel<<<BL / 256, 256, 0, stream>>>(gh, dw2[j], db2[j], gts, BL, DH);
    combine_kernel<<<ELT / 256, 256, 0, stream>>>(xin, p3b, p5b, p7b, gts, xout,
                                                  trend);
  };
  decomp(0, xc, xc, t1);

  // --- frequency-domain cross attention ------------------------------------
  gemm_wt_kernel<0, 0><<<dim3(Dd / 128, BL / 16), 256, 0, stream>>>(
      xc, wq, bq, nullptr, qb, BL, Dd, Dd);
  gemm_wt_kernel<0, 0><<<dim3(Dd / 128, BS / 16), 256, 0, stream>>>(
      cross, wk, bk, nullptr, kp, BS, Dd, Dd);
  gemm_wt_kernel<0, 0><<<dim3(Dd / 128, BS / 16), 256, 0, stream>>>(
      cross, wv, bv, nullptr, vp, BS, Dd, Dd);
  dft_fwd_kernel<<<dim3(1, Mm / 16, Bb * Hh), 128, 0, stream>>>(
      FcL, FsL, qb, Hh, DK, (long long)Ls * Dd, Dd, Ls, DK, qfr, qfi);
  dft_fwd_kernel<<<dim3(1, Mm / 16, Bb * Hh), 128, 0, stream>>>(
      FcS, FsS, kp, Hh, DK, (long long)Ss * Dd, Dd, Ss, DK, kfr, kfi);
  dft_fwd_kernel<<<dim3(1, Mm / 16, Bb * Hh), 128, 0, stream>>>(
      FcS, FsS, vp, Hh, DK, (long long)Ss * Dd, Dd, Ss, DK, vfr, vfi);
  attn_logits_kernel<<<dim3(1, 4, Bb * Hh), 128, 0, stream>>>(qfr, qfi, kfr,
                                                              kfi, att);
  softmax64_kernel<<<(Bb * Hh * 64) / 8, 256, 0, stream>>>(att, Bb * Hh * 64);
  attn_apply_kernel<<<dim3(1, 4, Bb * Hh), 128, 0, stream>>>(att, vfr, vfi,
                                                             ofr, ofi);
  irfft_kernel<<<dim3(1, Ls / 16, Bb * Hh), 128, 0, stream>>>(
      ILc, ILs, ofr, ofi, DK, Hh, qb, Dd, nullptr);        // qb reused as octx
  gemm_wt_kernel<0, 1><<<dim3(Dd / 128, BL / 16), 256, 0, stream>>>(
      qb, wo, bo, xc, xc, BL, Dd, Dd);                     // + residual

  decomp(1, xc, xc, t2);

  // --- FFN: x = x + relu(x @ w1^T) @ w2^T ----------------------------------
  gemm_wt_kernel<1, 0><<<dim3(DFFf / 128, BL / 16), 256, 0, stream>>>(
      xc, w_ff1, nullptr, nullptr, ffh, BL, DFFf, Dd);
  gemm_wt_kernel<0, 1><<<dim3(Dd / 128, BL / 16), 256, 0, stream>>>(
      ffh, w_ff2, nullptr, xc, xc, BL, Dd, DFFf);

  decomp(2, xc, out_x, t3);                                // seasonal -> out[0]

  // --- residual trend: sum of three circular convolutions -------------------
  circconv_kernel<<<dim3(COo / 128, Ls / 16, Bb), 256, 0, stream>>>(
      t1, t2, t3, pc1, pc2, pc3, out_tr);
}